// BahdanauAttentionLayer_11012296147168
// MI455X (gfx1250) — compile-verified
//
#include <hip/hip_runtime.h>
#include <cmath>

typedef __attribute__((ext_vector_type(2))) float v2f;
typedef __attribute__((ext_vector_type(8))) float v8f;

// Problem shape (fixed by the reference harness)
constexpr int B_  = 8;
constexpr int Te_ = 256;
constexpr int Td_ = 128;
constexpr int H_  = 512;   // also D (dec feature dim)

// ---------------------------------------------------------------------------
// Batched row-major GEMM via V_WMMA_F32_16X16X4_F32 with 2x2 register tiling.
//   C[b] (MxN) = A[b] (MxK) * Bm[b] (KxN)
// One wave32 computes a 32x32 output block (4 accumulators of 16x16).
// Per K-step (K+=4): 2 A frags + 2 B frags -> 4 WMMAs (2 dwords loaded/wmma,
// and each accumulator chain has 3 other WMMAs in between to hide the
// WMMA->WMMA RAW hazard).
// Fragment layout per CDNA5 ISA 7.12.2 (32-bit A 16x4 / B 4x16 / C 16x16):
//   A: vgpr0 = K=2*half, vgpr1 = K=2*half+1, row M = lane%16
//   B: vgpr0 = K=2*half, vgpr1 = K=2*half+1, col N = lane%16
//   D: vgpr i -> row (i + 8*half), col N = lane%16
// ---------------------------------------------------------------------------
__global__ __launch_bounds__(128)
void gemm_wmma_f32(const float* __restrict__ A,
                   const float* __restrict__ Bm,
                   float* __restrict__ C,
                   int M, int N, int K,
                   long sA, long sB, long sC)
{
    const int wave = threadIdx.x >> 5;
    const int lane = threadIdx.x & 31;
    const int half = lane >> 4;       // 0: lanes 0-15, 1: lanes 16-31
    const int l    = lane & 15;

    const int tilesN = N >> 5;        // 32-wide blocks
    const int tilesM = M >> 5;
    const int wid = blockIdx.x * (blockDim.x >> 5) + wave;
    if (wid >= tilesM * tilesN) return;
    const int tm = wid / tilesN;
    const int tn = wid - tm * tilesN;

    const long b = blockIdx.y;
    const float* Ar0 = A  + b * sA + (long)(tm * 32 + l) * K;       // rows 0..15
    const float* Ar1 = Ar0 + (long)16 * K;                          // rows 16..31
    const float* Bc0 = Bm + b * sB + tn * 32 + l;                   // cols 0..15
    const float* Bc1 = Bc0 + 16;                                    // cols 16..31

    v8f acc00 = {}, acc01 = {}, acc10 = {}, acc11 = {};

    #pragma unroll 4
    for (int k0 = 0; k0 < K; k0 += 4) {
        const int  ka = k0 + half * 2;
        const long r0 = (long)ka * N;
        const long r1 = (long)(ka + 1) * N;

        v2f a0; a0.x = Ar0[ka]; a0.y = Ar0[ka + 1];
        v2f a1; a1.x = Ar1[ka]; a1.y = Ar1[ka + 1];
        v2f b0; b0.x = Bc0[r0]; b0.y = Bc0[r1];
        v2f b1; b1.x = Bc1[r0]; b1.y = Bc1[r1];

        acc00 = __builtin_amdgcn_wmma_f32_16x16x4_f32(false, a0, false, b0,
                                                      (short)0, acc00, false, false);
        acc01 = __builtin_amdgcn_wmma_f32_16x16x4_f32(false, a0, false, b1,
                                                      (short)0, acc01, false, false);
        acc10 = __builtin_amdgcn_wmma_f32_16x16x4_f32(false, a1, false, b0,
                                                      (short)0, acc10, false, false);
        acc11 = __builtin_amdgcn_wmma_f32_16x16x4_f32(false, a1, false, b1,
                                                      (short)0, acc11, false, false);
    }

    // Store: row base for acc row-block i16 is tm*32 + i16*16 + half*8
    float* C00 = C + b * sC + (long)(tm * 32 + half * 8) * N + tn * 32 + l;
    float* C10 = C00 + (long)16 * N;
    #pragma unroll
    for (int i = 0; i < 8; ++i) {
        const long r = (long)i * N;
        C00[r]      = acc00[i];
        C00[r + 16] = acc01[i];
        C10[r]      = acc10[i];
        C10[r + 16] = acc11[i];
    }
}

// ---------------------------------------------------------------------------
// Fused score + softmax. One block per (b, d): 256 threads (8 wave32).
// Thread t computes scores[b,d,t] = sum_k tanh(enc_w[b,t,k]+dec_u[b,d,k])*v[k]
// (dec_u row and v cached in LDS, enc_w streamed as float4), then an LDS-tree
// max/sum softmax over Te=256, writing e[b,d,:].
// ---------------------------------------------------------------------------
__global__ __launch_bounds__(Te_)
void score_softmax(const float* __restrict__ enc_w,
                   const float* __restrict__ dec_u,
                   const float* __restrict__ v,
                   float* __restrict__ e_out)
{
    __shared__ __align__(16) float du[H_];
    __shared__ __align__(16) float vv[H_];
    __shared__ float red[Te_];

    const int bd = blockIdx.x;        // b*Td + d
    const int b  = bd / Td_;
    const int t  = threadIdx.x;       // encoder position, 0..Te-1

    for (int i = t; i < H_; i += Te_) {
        du[i] = dec_u[(long)bd * H_ + i];
        vv[i] = v[i];
    }
    __syncthreads();

    const float4* ew4 = reinterpret_cast<const float4*>(
        enc_w + ((long)b * Te_ + t) * H_);
    const float4* du4 = reinterpret_cast<const float4*>(du);
    const float4* vv4 = reinterpret_cast<const float4*>(vv);

    float s = 0.f;
    #pragma unroll 4
    for (int k = 0; k < H_ / 4; ++k) {
        const float4 a = ew4[k];
        const float4 d = du4[k];
        const float4 w = vv4[k];
        s += tanhf(a.x + d.x) * w.x;
        s += tanhf(a.y + d.y) * w.y;
        s += tanhf(a.z + d.z) * w.z;
        s += tanhf(a.w + d.w) * w.w;
    }

    // max over Te
    red[t] = s;
    __syncthreads();
    for (int off = Te_ / 2; off > 0; off >>= 1) {
        if (t < off) red[t] = fmaxf(red[t], red[t + off]);
        __syncthreads();
    }
    const float m = red[0];
    __syncthreads();

    // sum of exp
    const float ex = __expf(s - m);
    red[t] = ex;
    __syncthreads();
    for (int off = Te_ / 2; off > 0; off >>= 1) {
        if (t < off) red[t] += red[t + off];
        __syncthreads();
    }
    const float inv = 1.f / red[0];

    e_out[(long)bd * Te_ + t] = ex * inv;
}

extern "C" void kernel_launch(void* const* d_in, const int* in_sizes, int n_in,
                              void* d_out, int out_size, void* d_ws, size_t ws_size,
                              hipStream_t stream) {
    const float* enc_out = (const float*)d_in[0];  // [8,256,512]
    const float* dec_out = (const float*)d_in[1];  // [8,128,512]
    const float* W       = (const float*)d_in[2];  // [512,512]
    const float* U       = (const float*)d_in[3];  // [512,512]
    const float* v       = (const float*)d_in[4];  // [512,1]

    float* out   = (float*)d_out;
    float* c_out = out;                                  // [8,128,512]
    float* e_out = out + (long)B_ * Td_ * H_;            // [8,128,256]

    float* enc_w = (float*)d_ws;                         // [8*256, 512] = 4 MB
    float* dec_u = enc_w + (long)B_ * Te_ * H_;          // [8*128, 512] = 2 MB

    // 1) enc_w = enc_out @ W   (M=2048, N=512, K=512, batch=1)
    {
        const int M = B_ * Te_, N = H_, K = H_;
        const int tiles = (M / 32) * (N / 32);
        gemm_wmma_f32<<<dim3((tiles + 3) / 4, 1), 128, 0, stream>>>(
            enc_out, W, enc_w, M, N, K, 0, 0, 0);
    }
    // 2) dec_u = dec_out @ U   (M=1024, N=512, K=512, batch=1)
    {
        const int M = B_ * Td_, N = H_, K = H_;
        const int tiles = (M / 32) * (N / 32);
        gemm_wmma_f32<<<dim3((tiles + 3) / 4, 1), 128, 0, stream>>>(
            dec_out, U, dec_u, M, N, K, 0, 0, 0);
    }
    // 3) e = softmax_t( tanh(enc_w + dec_u) . v )   -> e_out
    score_softmax<<<B_ * Td_, Te_, 0, stream>>>(enc_w, dec_u, v, e_out);

    // 4) c[b] = e[b] (128x256) @ enc_out[b] (256x512)   batched over b
    {
        const int M = Td_, N = H_, K = Te_;
        const int tiles = (M / 32) * (N / 32);
        gemm_wmma_f32<<<dim3((tiles + 3) / 4, B_), 128, 0, stream>>>(
            e_out, enc_out, c_out, M, N, K,
            (long)Td_ * Te_, (long)Te_ * H_, (long)Td_ * H_);
    }
}